// Local3dAttention_54528904790695
// MI455X (gfx1250) — compile-verified
//
#include <hip/hip_runtime.h>
#include <cmath>

// ---- problem constants (from reference) ----
#define DIM    256
#define INNER  256
#define HEADS  8
#define DH     32
#define B_     2
#define S_     8
#define H_     32
#define W_     16
#define NTOK   (B_*S_*H_*W_)      // 8192 tokens
#define NKEY   300                 // 3*5*20 widened key window per w-row
#define NKEYP  304                 // padded to 19 tiles of 16
#define SCALE  0.17677669529663687f // 1/sqrt(32)
#define MASKV  (-1e9f)

typedef __attribute__((ext_vector_type(2))) float v2f;
typedef __attribute__((ext_vector_type(4))) float v4f;
typedef __attribute__((ext_vector_type(8))) float v8f;

// CDNA5 async global->LDS copy (ASYNCcnt-tracked), 16B per lane.
// LDS offset = low 32 bits of the generic shared-memory address.
__device__ inline void async_copy_b128(float* lds_dst, const float* gsrc)
{
    unsigned loff = (unsigned)(uintptr_t)lds_dst;
    asm volatile("global_load_async_to_lds_b128 %0, %1, off"
                 :: "v"(loff), "v"(gsrc) : "memory");
}
__device__ inline void wait_asynccnt0()
{
    asm volatile("s_wait_asynccnt 0x0" ::: "memory");
}

// =====================================================================
// Kernel 1: fused Q/K/V projections.  out[n,j] = sum_d src[n,d]*W[j,d]
// One wave computes a 16x64 tile (4 accumulators) -> A regs reused 4x.
// =====================================================================
__global__ void __launch_bounds__(32) proj_kernel(
    const float* __restrict__ x, const float* __restrict__ q,
    const float* __restrict__ wq, const float* __restrict__ wk,
    const float* __restrict__ wv,
    float* __restrict__ qp, float* __restrict__ kb, float* __restrict__ vb)
{
    const int lane = threadIdx.x & 31;
    const int half = lane >> 4;        // K-pair selector per ISA A/B layout
    const int l16  = lane & 15;
    const int row0 = blockIdx.x * 16;
    const int col0 = blockIdx.y * 64;
    const int mat  = blockIdx.z;

    const float* src = (mat == 0) ? q  : x;
    const float* wt  = (mat == 0) ? wq : (mat == 1) ? wk : wv;
    float*       dst = (mat == 0) ? qp : (mat == 1) ? kb : vb;

    const float* arow = src + (size_t)(row0 + l16) * DIM;     // A: M=l16
    const float* br0  = wt  + (size_t)(col0 +  0 + l16) * DIM; // B[k][n]=W[n][k]
    const float* br1  = wt  + (size_t)(col0 + 16 + l16) * DIM;
    const float* br2  = wt  + (size_t)(col0 + 32 + l16) * DIM;
    const float* br3  = wt  + (size_t)(col0 + 48 + l16) * DIM;

    v8f c0 = {}, c1 = {}, c2 = {}, c3 = {};
#pragma unroll 4
    for (int k0 = 0; k0 < DIM; k0 += 4) {
        int ko = k0 + 2 * half;
        v2f a  = *(const v2f*)(arow + ko);
        v2f b0 = *(const v2f*)(br0 + ko);
        v2f b1 = *(const v2f*)(br1 + ko);
        v2f b2 = *(const v2f*)(br2 + ko);
        v2f b3 = *(const v2f*)(br3 + ko);
        c0 = __builtin_amdgcn_wmma_f32_16x16x4_f32(false, a, false, b0, (short)0, c0, false, false);
        c1 = __builtin_amdgcn_wmma_f32_16x16x4_f32(false, a, false, b1, (short)0, c1, false, false);
        c2 = __builtin_amdgcn_wmma_f32_16x16x4_f32(false, a, false, b2, (short)0, c2, false, false);
        c3 = __builtin_amdgcn_wmma_f32_16x16x4_f32(false, a, false, b3, (short)0, c3, false, false);
    }
#pragma unroll
    for (int i = 0; i < 8; ++i) {
        int m = i + 8 * half;          // C/D: lane half selects M 0..7 / 8..15
        float* orow = dst + (size_t)(row0 + m) * INNER + l16;
        orow[col0 +  0] = c0[i];
        orow[col0 + 16] = c1[i];
        orow[col0 + 32] = c2[i];
        orow[col0 + 48] = c3[i];
    }
}

// =====================================================================
// Kernel 2: local attention for one (b,s,h) row (16 queries) x one head.
// Async-staged K/V window (304x32 each), WMMA dots (16x32 @ 32x304),
// masked softmax, K-split WMMA attn@V (16x304 @ 304x32) over all 8 waves.
// =====================================================================
__global__ void __launch_bounds__(256) attn_kernel(
    const float* __restrict__ qp, const float* __restrict__ kbuf,
    const float* __restrict__ vbuf, float* __restrict__ attn_out)
{
    extern __shared__ float smem[];
    float* Ks   = smem;                   // NKEYP*DH
    float* Vs   = Ks + NKEYP * DH;        // NKEYP*DH
    float* Qs   = Vs + NKEYP * DH;        // 16*DH
    float* Pr   = Qs + 16 * DH;           // 16*NKEYP  (dots -> probs)
    float* Red  = Pr + 16 * NKEYP;        // 16*16 reduction scratch
    float* RowV = Red + 256;              // 16 row max / inv-sum
    float* Acc  = RowV + 16;              // 8*256 attn@V partials

    const int tid  = threadIdx.x;
    const int wave = tid >> 5;
    const int lane = tid & 31;
    const int half = lane >> 4;
    const int l16  = lane & 15;

    const int bsh  = blockIdx.x;          // (b,s,h) flat index
    const int head = blockIdx.y;
    const int b_i  = bsh / (S_ * H_);
    const int rem  = bsh % (S_ * H_);
    const int s_i  = rem / H_;
    const int h_i  = rem % H_;
    const int nbase = bsh * W_;           // first token of this w-row
    const int hoff  = head * DH;

    // ---- async-stage Q tile (16 x 32), 16B chunks ----
    for (int idx = tid; idx < 16 * (DH / 4); idx += 256) {
        int m = idx >> 3, d4 = (idx & 7) * 4;
        async_copy_b128(Qs + m * DH + d4,
                        qp + (size_t)(nbase + m) * INNER + hoff + d4);
    }
    // ---- stage K/V window: async b128 for valid rows, zero-fill rest ----
    for (int idx = tid; idx < NKEYP * (DH / 4); idx += 256) {
        int key = idx >> 3, d4 = (idx & 7) * 4;
        bool   valid = false;
        size_t g     = 0;
        if (key < NKEY) {
            int ds  = key / 100 - 1;
            int r2  = key % 100;
            int dhh = r2 / 20 - 2;
            int ww  = r2 % 20 - 2;
            int ss = s_i + ds, hh = h_i + dhh;
            if (ss >= 0 && ss < S_ && hh >= 0 && hh < H_ && ww >= 0 && ww < W_) {
                valid = true;
                g = (size_t)((((b_i * S_ + ss) * H_ + hh) * W_) + ww) * INNER
                  + hoff + d4;
            }
        }
        if (valid) {
            async_copy_b128(Ks + key * DH + d4, kbuf + g);
            async_copy_b128(Vs + key * DH + d4, vbuf + g);
        } else {
            v4f z = {};
            *(v4f*)(Ks + key * DH + d4) = z;
            *(v4f*)(Vs + key * DH + d4) = z;
        }
    }
    wait_asynccnt0();          // this wave's async copies landed in LDS
    __syncthreads();           // all waves' copies visible

    // ---- dots = Q @ K^T, 19 key tiles distributed over 8 waves ----
    for (int tile = wave; tile < 19; tile += 8) {
        v8f c = {};
        const float* aa = Qs + l16 * DH;                 // A: query row l16
        const float* bb = Ks + (tile * 16 + l16) * DH;   // B[k][n]=Ks[key][k]
#pragma unroll
        for (int k0 = 0; k0 < DH; k0 += 4) {
            v2f a = *(const v2f*)(aa + k0 + 2 * half);
            v2f b = *(const v2f*)(bb + k0 + 2 * half);
            c = __builtin_amdgcn_wmma_f32_16x16x4_f32(false, a, false, b,
                                                      (short)0, c, false, false);
        }
        // per-(query,key) mask, scale, write to LDS probs
        int  key    = tile * 16 + l16;
        bool gvalid = false;
        int  kwi    = 0;
        if (key < NKEY) {
            int ds  = key / 100 - 1;
            int r2  = key % 100;
            int dhh = r2 / 20 - 2;
            kwi     = r2 % 20;
            int ss = s_i + ds, hh = h_i + dhh, ww = kwi - 2;
            gvalid = (ss >= 0 && ss < S_ && hh >= 0 && hh < H_ &&
                      ww >= 0 && ww < W_);
        }
#pragma unroll
        for (int i = 0; i < 8; ++i) {
            int  m  = i + 8 * half;                      // query w position
            bool ok = gvalid && (kwi >= m) && (kwi <= m + 4); // |ww - m| <= 2
            Pr[m * NKEYP + key] = ok ? c[i] * SCALE : MASKV;
        }
    }
    __syncthreads();

    // ---- softmax per query row (16 threads per row over 304 cols) ----
    {
        int row = tid >> 4, sub = tid & 15;
        float mx = -INFINITY;
        for (int col = sub; col < NKEYP; col += 16)
            mx = fmaxf(mx, Pr[row * NKEYP + col]);
        Red[row * 16 + sub] = mx;
        __syncthreads();
        if (sub == 0) {
            float m2 = Red[row * 16];
            for (int j = 1; j < 16; ++j) m2 = fmaxf(m2, Red[row * 16 + j]);
            RowV[row] = m2;
        }
        __syncthreads();
        float rmax = RowV[row];
        float sum  = 0.f;
        for (int col = sub; col < NKEYP; col += 16) {
            float e = __expf(Pr[row * NKEYP + col] - rmax);
            Pr[row * NKEYP + col] = e;
            sum += e;
        }
        Red[row * 16 + sub] = sum;
        __syncthreads();
        if (sub == 0) {
            float s2 = 0.f;
            for (int j = 0; j < 16; ++j) s2 += Red[row * 16 + j];
            RowV[row] = 1.f / s2;
        }
        __syncthreads();
        float inv = RowV[row];
        for (int col = sub; col < NKEYP; col += 16)
            Pr[row * NKEYP + col] *= inv;
    }
    __syncthreads();

    // ---- out = attn @ V, K-split across 8 waves (2 N-tiles x 4 K-slices) ----
    {
        int ntile  = wave >> 2;            // 0..1 -> dh columns 0..15 / 16..31
        int kslice = wave & 3;             // 0..3 -> 76 keys each, 19 K-chunks of 4 keys
        int nb     = ntile * 16;
        v8f c = {};
        for (int kc = kslice * 76; kc < (kslice + 1) * 76; kc += 4) {
            v2f a = *(const v2f*)(Pr + l16 * NKEYP + kc + 2 * half);
            v2f b;
            b.x = Vs[(kc + 2 * half)     * DH + nb + l16];
            b.y = Vs[(kc + 2 * half + 1) * DH + nb + l16];
            c = __builtin_amdgcn_wmma_f32_16x16x4_f32(false, a, false, b,
                                                      (short)0, c, false, false);
        }
#pragma unroll
        for (int i = 0; i < 8; ++i)
            Acc[wave * 256 + i * 32 + lane] = c[i];
    }
    __syncthreads();

    // ---- reduce K-slices and store head output (16 x 32) ----
    for (int t = tid; t < 16 * DH; t += 256) {
        int m = t >> 5, n = t & 31;
        int ntile = n >> 4;
        int slot  = (m & 7) * 32 + (n & 15) + ((m >= 8) ? 16 : 0);
        float s = 0.f;
#pragma unroll
        for (int ks = 0; ks < 4; ++ks)
            s += Acc[(ntile * 4 + ks) * 256 + slot];
        attn_out[(size_t)(nbase + m) * INNER + hoff + n] = s;
    }
}

// =====================================================================
// Kernel 3: output projection + bias, 16x64 tile per wave.
// =====================================================================
__global__ void __launch_bounds__(32) outproj_kernel(
    const float* __restrict__ a, const float* __restrict__ wo,
    const float* __restrict__ bo, float* __restrict__ out)
{
    const int lane = threadIdx.x & 31;
    const int half = lane >> 4;
    const int l16  = lane & 15;
    const int row0 = blockIdx.x * 16;
    const int col0 = blockIdx.y * 64;

    const float* arow = a  + (size_t)(row0 + l16) * INNER;
    const float* br0  = wo + (size_t)(col0 +  0 + l16) * INNER;
    const float* br1  = wo + (size_t)(col0 + 16 + l16) * INNER;
    const float* br2  = wo + (size_t)(col0 + 32 + l16) * INNER;
    const float* br3  = wo + (size_t)(col0 + 48 + l16) * INNER;

    v8f c0 = {}, c1 = {}, c2 = {}, c3 = {};
#pragma unroll 4
    for (int k0 = 0; k0 < INNER; k0 += 4) {
        int ko = k0 + 2 * half;
        v2f av = *(const v2f*)(arow + ko);
        v2f b0 = *(const v2f*)(br0 + ko);
        v2f b1 = *(const v2f*)(br1 + ko);
        v2f b2 = *(const v2f*)(br2 + ko);
        v2f b3 = *(const v2f*)(br3 + ko);
        c0 = __builtin_amdgcn_wmma_f32_16x16x4_f32(false, av, false, b0, (short)0, c0, false, false);
        c1 = __builtin_amdgcn_wmma_f32_16x16x4_f32(false, av, false, b1, (short)0, c1, false, false);
        c2 = __builtin_amdgcn_wmma_f32_16x16x4_f32(false, av, false, b2, (short)0, c2, false, false);
        c3 = __builtin_amdgcn_wmma_f32_16x16x4_f32(false, av, false, b3, (short)0, c3, false, false);
    }
#pragma unroll
    for (int i = 0; i < 8; ++i) {
        int m = i + 8 * half;
        float* orow = out + (size_t)(row0 + m) * DIM + l16;
        orow[col0 +  0] = c0[i] + bo[col0 +  0 + l16];
        orow[col0 + 16] = c1[i] + bo[col0 + 16 + l16];
        orow[col0 + 32] = c2[i] + bo[col0 + 32 + l16];
        orow[col0 + 48] = c3[i] + bo[col0 + 48 + l16];
    }
}

extern "C" void kernel_launch(void* const* d_in, const int* in_sizes, int n_in,
                              void* d_out, int out_size, void* d_ws, size_t ws_size,
                              hipStream_t stream)
{
    const float* x  = (const float*)d_in[0];
    const float* q  = (const float*)d_in[1];
    const float* wq = (const float*)d_in[2];
    const float* wk = (const float*)d_in[3];
    const float* wv = (const float*)d_in[4];
    const float* wo = (const float*)d_in[5];
    const float* bo = (const float*)d_in[6];
    float* out = (float*)d_out;

    // workspace layout: qp | k | v | attn_out  (4 x 8 MB, all L2-resident)
    float* ws = (float*)d_ws;
    float* qp = ws;
    float* kb = qp + (size_t)NTOK * INNER;
    float* vb = kb + (size_t)NTOK * INNER;
    float* ao = vb + (size_t)NTOK * INNER;

    proj_kernel<<<dim3(NTOK / 16, INNER / 64, 3), 32, 0, stream>>>(
        x, q, wq, wk, wv, qp, kb, vb);

    size_t smem = (size_t)(2 * NKEYP * DH + 16 * DH + 16 * NKEYP + 256 + 16
                           + 8 * 256) * sizeof(float); // ~107 KB < 320 KB WGP LDS
    attn_kernel<<<dim3(B_ * S_ * H_, HEADS), 256, smem, stream>>>(
        qp, kb, vb, ao);

    outproj_kernel<<<dim3(NTOK / 16, DIM / 64), 32, 0, stream>>>(
        ao, wo, bo, out);
}